// MeshGNN_1838246002766
// MI455X (gfx1250) — compile-verified
//
#include <hip/hip_runtime.h>
#include <hip/hip_bf16.h>

typedef float v2f __attribute__((ext_vector_type(2)));
typedef float v8f __attribute__((ext_vector_type(8)));

#define IN_F 64
#define HID_F 64
#define OUT_F 32

// ---------------------------------------------------------------------------
// Utility: zero-fill a float buffer
// ---------------------------------------------------------------------------
__global__ void fill_zero_kernel(float* __restrict__ p, long long n) {
    long long i = (long long)blockIdx.x * blockDim.x + threadIdx.x;
    if (i < n) p[i] = 0.0f;
}

// ---------------------------------------------------------------------------
// Degree: deg[dst[e]] += 1  (deg buffer pre-zeroed)
// ---------------------------------------------------------------------------
__global__ void degree_kernel(const int* __restrict__ dst, float* __restrict__ deg,
                              int nedges) {
    int e = blockIdx.x * blockDim.x + threadIdx.x;
    if (e < nedges) atomicAdd(&deg[dst[e]], 1.0f);
}

// dinv[i] = rsqrt(deg[i] + 1)  (in place: deg buffer becomes dinv)
__global__ void rsqrt_kernel(float* __restrict__ d, int n) {
    int i = blockIdx.x * blockDim.x + threadIdx.x;
    if (i < n) d[i] = rsqrtf(d[i] + 1.0f);
}

// ---------------------------------------------------------------------------
// H = X @ W  with V_WMMA_F32_16X16X4_F32.
//   X: [nrows x 64] row-major, W: [64 x OC] row-major, H: [nrows x OC].
// One wave32 computes a 16-row strip across all OC columns (OC/16 tiles),
// looping K=64 in steps of 4. Fragment layouts per CDNA5 ISA 7.12.2:
//   A 16x4 : lanes 0-15 -> {K=k0, k0+1}, lanes 16-31 -> {K=k0+2, k0+3}, M=lane&15
//   B 4x16 : lanes 0-15 -> rows {k0, k0+1}, lanes 16-31 -> rows {k0+2, k0+3}, N=lane&15
//   D      : VGPR v -> row (v + 8*(lane>=16)), col = lane&15
// nrows must be a multiple of 16 (N=100000 is).
// ---------------------------------------------------------------------------
template <int OC>
__global__ __launch_bounds__(128) void gemm_xw_wmma(const float* __restrict__ X,
                                                    const float* __restrict__ W,
                                                    float* __restrict__ H,
                                                    int nrows) {
    const int lane  = threadIdx.x & 31;
    const int wave  = threadIdx.x >> 5;
    const int strip = blockIdx.x * (blockDim.x >> 5) + wave;
    const int row_base = strip * 16;
    if (row_base >= nrows) return;  // uniform per wave -> EXEC stays all-ones

    const int lhalf = lane & 15;        // M (for A) / N (for B, D)
    const int khi   = (lane >> 4) * 2;  // 0 for lanes 0-15, 2 for lanes 16-31
    const int mofs  = (lane >> 4) * 8;  // D-row offset for upper half-wave

    constexpr int NT = OC / 16;
    v8f acc[NT];
#pragma unroll
    for (int t = 0; t < NT; ++t)
#pragma unroll
        for (int v = 0; v < 8; ++v) acc[t][v] = 0.0f;

    const float* xrow = X + (size_t)(row_base + lhalf) * IN_F;
#pragma unroll
    for (int k0 = 0; k0 < 64; k0 += 4) {
        v2f a;
        a.x = xrow[k0 + khi + 0];
        a.y = xrow[k0 + khi + 1];
#pragma unroll
        for (int t = 0; t < NT; ++t) {
            const float* wp = W + (size_t)(k0 + khi) * OC + t * 16 + lhalf;
            v2f b;
            b.x = wp[0];
            b.y = wp[OC];
            acc[t] = __builtin_amdgcn_wmma_f32_16x16x4_f32(
                false, a, false, b, (short)0, acc[t], false, false);
        }
    }

#pragma unroll
    for (int t = 0; t < NT; ++t) {
        float* hp = H + (size_t)row_base * OC + t * 16 + lhalf;
#pragma unroll
        for (int v = 0; v < 8; ++v) {
            hp[(size_t)(v + mofs) * OC] = acc[t][v];
        }
    }
}

// ---------------------------------------------------------------------------
// Edge scatter: agg[dst] += h[src] * dinv[src]*dinv[dst]
// F/4 threads per edge; each thread gathers a float4 (B128 load) and issues
// 4 global_atomic_add_f32.
// ---------------------------------------------------------------------------
template <int F>
__global__ __launch_bounds__(256) void scatter_edges(const float* __restrict__ h,
                                                     const float* __restrict__ dinv,
                                                     const int* __restrict__ src,
                                                     const int* __restrict__ dst,
                                                     float* __restrict__ agg,
                                                     int nedges) {
    constexpr int TPE = F / 4;
    long long gid = (long long)blockIdx.x * blockDim.x + threadIdx.x;
    int e  = (int)(gid / TPE);
    int fo = (int)(gid % TPE) * 4;
    if (e >= nedges) return;
    int s = src[e];
    int d = dst[e];
    float norm = dinv[s] * dinv[d];
    const float4 hv = *(const float4*)(h + (size_t)s * F + fo);
    float* ap = agg + (size_t)d * F + fo;
    atomicAdd(ap + 0, hv.x * norm);
    atomicAdd(ap + 1, hv.y * norm);
    atomicAdd(ap + 2, hv.z * norm);
    atomicAdd(ap + 3, hv.w * norm);
}

// ---------------------------------------------------------------------------
// Layer-1 finalize (in place on agg): agg = relu(agg + hw*dinv^2 + b)
// ---------------------------------------------------------------------------
__global__ void finalize_relu_kernel(const float* __restrict__ hw,
                                     const float* __restrict__ dinv,
                                     const float* __restrict__ b,
                                     float* __restrict__ agg, int n) {
    long long gid = (long long)blockIdx.x * blockDim.x + threadIdx.x;
    long long total = (long long)n * HID_F;
    if (gid >= total) return;
    int i = (int)(gid / HID_F);
    int f = (int)(gid % HID_F);
    float di = dinv[i];
    float v  = agg[gid] + hw[gid] * di * di + b[f];
    agg[gid] = fmaxf(v, 0.0f);
}

// ---------------------------------------------------------------------------
// Layer-2 finalize: out += hw*dinv^2 + b   (out already holds edge scatter sum)
// ---------------------------------------------------------------------------
__global__ void finalize_out_kernel(const float* __restrict__ hw,
                                    const float* __restrict__ dinv,
                                    const float* __restrict__ b,
                                    float* __restrict__ out, int n) {
    long long gid = (long long)blockIdx.x * blockDim.x + threadIdx.x;
    long long total = (long long)n * OUT_F;
    if (gid >= total) return;
    int i = (int)(gid / OUT_F);
    int f = (int)(gid % OUT_F);
    float di = dinv[i];
    out[gid] = out[gid] + hw[gid] * di * di + b[f];
}

// ---------------------------------------------------------------------------
// Launch
// ---------------------------------------------------------------------------
extern "C" void kernel_launch(void* const* d_in, const int* in_sizes, int n_in,
                              void* d_out, int out_size, void* d_ws, size_t ws_size,
                              hipStream_t stream) {
    const float* x  = (const float*)d_in[0];
    const float* W1 = (const float*)d_in[1];
    const float* b1 = (const float*)d_in[2];
    const float* W2 = (const float*)d_in[3];
    const float* b2 = (const float*)d_in[4];
    const int*   ei = (const int*)d_in[5];

    const int N = in_sizes[0] / IN_F;     // 100000
    const int E = in_sizes[5] / 2;        // 1280000
    const int* src = ei;
    const int* dst = ei + E;

    // Workspace layout: dinv[N] | H[N*64] | AGG[N*64]   (~51.6 MB)
    float* dinv = (float*)d_ws;
    float* Hbuf = dinv + N;
    float* Abuf = Hbuf + (size_t)N * HID_F;
    float* out  = (float*)d_out;

    const int T = 256;
    long long nH = (long long)N * HID_F;
    long long nO = (long long)N * OUT_F;

    // 1) zero scratch + output (harness poisons; we need clean accumulators)
    fill_zero_kernel<<<(unsigned)((N + T - 1) / T), T, 0, stream>>>(dinv, N);
    fill_zero_kernel<<<(unsigned)((nH + T - 1) / T), T, 0, stream>>>(Abuf, nH);
    fill_zero_kernel<<<(unsigned)((nO + T - 1) / T), T, 0, stream>>>(out, nO);

    // 2) degree -> dinv
    degree_kernel<<<(unsigned)((E + T - 1) / T), T, 0, stream>>>(dst, dinv, E);
    rsqrt_kernel<<<(unsigned)((N + T - 1) / T), T, 0, stream>>>(dinv, N);

    // 3) H1 = x @ W1   (WMMA f32 16x16x4)
    {
        int strips = N / 16;                 // 6250, exact
        int waves_per_block = 4;             // 128 threads
        int blocks = (strips + waves_per_block - 1) / waves_per_block;
        gemm_xw_wmma<HID_F><<<blocks, 128, 0, stream>>>(x, W1, Hbuf, N);
    }

    // 4) layer-1 edge scatter into Abuf
    {
        long long thr = (long long)E * (HID_F / 4);
        scatter_edges<HID_F><<<(unsigned)((thr + T - 1) / T), T, 0, stream>>>(
            Hbuf, dinv, src, dst, Abuf, E);
    }

    // 5) self-loop + bias + ReLU (Abuf becomes h)
    finalize_relu_kernel<<<(unsigned)((nH + T - 1) / T), T, 0, stream>>>(
        Hbuf, dinv, b1, Abuf, N);

    // 6) H2 = h @ W2   (reuse Hbuf as [N x 32])
    {
        int strips = N / 16;
        int blocks = (strips + 3) / 4;
        gemm_xw_wmma<OUT_F><<<blocks, 128, 0, stream>>>(Abuf, W2, Hbuf, N);
    }

    // 7) layer-2 edge scatter directly into out
    {
        long long thr = (long long)E * (OUT_F / 4);
        scatter_edges<OUT_F><<<(unsigned)((thr + T - 1) / T), T, 0, stream>>>(
            Hbuf, dinv, src, dst, out, E);
    }

    // 8) final self-loop + bias
    finalize_out_kernel<<<(unsigned)((nO + T - 1) / T), T, 0, stream>>>(
        Hbuf, dinv, b2, out, N);
}